// GAE_87771951661298
// MI455X (gfx1250) — compile-verified
//
#include <hip/hip_runtime.h>

// ============================================================================
// GAE (2x GATConv -> Linear -> mean-pool; MLP decoder) for MI455X / gfx1250.
// Dense GEMMs run through V_WMMA_F32_16X16X4_F32 (full fp32: workload is
// bandwidth/atomics bound, so no precision downgrade is needed).
// GEMM: 8 waves/block, each wave -> 16x64 output tile (4 WMMA accumulators),
// B panel (<=32KB) staged once per block in LDS (transposed + padded).
// ============================================================================

typedef float v2f __attribute__((ext_vector_type(2)));
typedef float v8f __attribute__((ext_vector_type(8)));

#define WAVE 32
#define BN   64            // N-panel width per block (N % 64 == 0 here)
#define KMAX 128           // max K over all GEMMs in this model
#define BPAD 2             // LDS row padding (floats)

// ---------------------------------------------------------------------------
// C[M,N] = act(A[M,K] @ B[K,N] + bias[N]); row-major; M%16==0, N%64==0, K%4==0.
// fp32 WMMA operand layout (ISA 7.12.2, 16x4 A / 4x16 B / 16x16 C):
//   A: lane (L&15)=row; lanes 0-15 carry K=k,k+1; lanes 16-31 carry K=k+2,k+3
//   B: lane (L&15)=col; same K striping
//   C: VGPR j -> row tile*16 + j (lanes 0-15) / +8+j (lanes 16-31)
// ---------------------------------------------------------------------------
__global__ void gemm_wmma_f32(const float* __restrict__ A,
                              const float* __restrict__ B,
                              const float* __restrict__ bias,
                              float* __restrict__ C,
                              int M, int K, int N, int act)
{
    __shared__ float Bs[BN][KMAX + BPAD];        // transposed panel: Bs[n][k]

    const int ngroups = N / BN;
    const int mblk = blockIdx.x / ngroups;       // block of 8 consecutive M-tiles
    const int ng   = blockIdx.x - mblk * ngroups;
    const int n0   = ng * BN;

    // ---- stage B panel: B[k][n0+n] -> Bs[n][k] (global reads coalesced in n)
    for (int idx = threadIdx.x; idx < K * BN; idx += (int)blockDim.x) {
        const int k = idx >> 6;                  // idx / 64
        const int n = idx & (BN - 1);
        Bs[n][k] = B[(size_t)k * N + n0 + n];
    }
    __syncthreads();

    const int wv   = threadIdx.x / WAVE;         // 0..7
    const int lane = threadIdx.x & (WAVE - 1);
    const int tm   = mblk * 8 + wv;              // 16-row tile index
    if (tm * 16 >= M) return;                    // wave-uniform (after barrier)

    const int l15   = lane & 15;
    const int khalf = (lane >> 4) << 1;          // 0 (lanes 0-15) / 2 (16-31)
    const int row   = tm * 16 + l15;
    const float* ap = A + (size_t)row * K + khalf;

    v8f acc0 = {}, acc1 = {}, acc2 = {}, acc3 = {};
    for (int k = 0; k < K; k += 4) {
        v2f a;
        a.x = ap[k];
        a.y = ap[k + 1];
        const int kk = k + khalf;
        v2f b0, b1, b2, b3;
        b0.x = Bs[l15     ][kk]; b0.y = Bs[l15     ][kk + 1];
        b1.x = Bs[l15 + 16][kk]; b1.y = Bs[l15 + 16][kk + 1];
        b2.x = Bs[l15 + 32][kk]; b2.y = Bs[l15 + 32][kk + 1];
        b3.x = Bs[l15 + 48][kk]; b3.y = Bs[l15 + 48][kk + 1];
        acc0 = __builtin_amdgcn_wmma_f32_16x16x4_f32(false, a, false, b0, (short)0, acc0, false, false);
        acc1 = __builtin_amdgcn_wmma_f32_16x16x4_f32(false, a, false, b1, (short)0, acc1, false, false);
        acc2 = __builtin_amdgcn_wmma_f32_16x16x4_f32(false, a, false, b2, (short)0, acc2, false, false);
        acc3 = __builtin_amdgcn_wmma_f32_16x16x4_f32(false, a, false, b3, (short)0, acc3, false, false);
    }

    const int rbase = tm * 16 + ((lane >> 4) << 3);      // +8 for lanes 16-31
    v8f accs[4] = {acc0, acc1, acc2, acc3};
#pragma unroll
    for (int q = 0; q < 4; ++q) {
        const int col = n0 + q * 16 + l15;
        const float bv = bias ? bias[col] : 0.0f;
        float* cp = C + (size_t)rbase * N + col;
#pragma unroll
        for (int j = 0; j < 8; ++j) {
            float v = accs[q][j] + bv;
            if (act) v = fmaxf(v, 0.0f);
            cp[(size_t)j * N] = v;
        }
    }
}

// ---------------------------------------------------------------------------
// Per-node attention logits: as[i] = h[i].a_src ; ad[i] = h[i].a_dst
// ---------------------------------------------------------------------------
__global__ void node_alpha(const float* __restrict__ h,
                           const float* __restrict__ avs,
                           const float* __restrict__ avd,
                           float* __restrict__ as, float* __restrict__ ad,
                           int n, int D)
{
    const int wid  = blockIdx.x * (blockDim.x / WAVE) + threadIdx.x / WAVE;
    const int lane = threadIdx.x & (WAVE - 1);
    if (wid >= n) return;
    const float* hp = h + (size_t)wid * D;
    float s0 = 0.0f, s1 = 0.0f;
    for (int c = lane; c < D; c += WAVE) {
        const float v = hp[c];
        s0 += v * avs[c];
        s1 += v * avd[c];
    }
    for (int off = WAVE / 2; off > 0; off >>= 1) {
        s0 += __shfl_down(s0, off, WAVE);
        s1 += __shfl_down(s1, off, WAVE);
    }
    if (lane == 0) { as[wid] = s0; ad[wid] = s1; }
}

// ---------------------------------------------------------------------------
// Elementwise helpers
// ---------------------------------------------------------------------------
__global__ void init_nodes(float* __restrict__ nmax, float* __restrict__ den, int n)
{
    const int i = blockIdx.x * blockDim.x + threadIdx.x;
    if (i < n) { nmax[i] = -__builtin_inff(); den[i] = 0.0f; }
}

__global__ void bcast_bias(float* __restrict__ out, const float* __restrict__ bias,
                           int total, int D)
{
    const int i = blockIdx.x * blockDim.x + threadIdx.x;
    if (i < total) out[i] = bias[i % D];
}

__global__ void zero_f(float* __restrict__ p, int total)
{
    const int i = blockIdx.x * blockDim.x + threadIdx.x;
    if (i < total) p[i] = 0.0f;
}

__global__ void relu_ip(float* __restrict__ p, int total)
{
    const int i = blockIdx.x * blockDim.x + threadIdx.x;
    if (i < total) p[i] = fmaxf(p[i], 0.0f);
}

// Float atomic-max via monotone int reinterpretation.
__device__ __forceinline__ void atomicMaxF(float* addr, float v)
{
    if (v >= 0.0f) atomicMax((int*)addr, __float_as_int(v));
    else           atomicMin((unsigned int*)addr, __float_as_uint(v));
}

// ---------------------------------------------------------------------------
// Edge passes. Edges e < nE come from edge_index; e >= nE are the implicit
// self-loops (src = dst = e - nE), matching GATConv add_self_loops=True.
// ---------------------------------------------------------------------------
__global__ void edge_max(const int* __restrict__ src, const int* __restrict__ dst,
                         const float* __restrict__ as, const float* __restrict__ ad,
                         float* __restrict__ esc, float* __restrict__ nmax,
                         int nE, int nNodes)
{
    const int e = blockIdx.x * blockDim.x + threadIdx.x;
    if (e >= nE + nNodes) return;
    int s, d;
    if (e < nE) { s = src[e]; d = dst[e]; } else { s = e - nE; d = s; }
    float v = as[s] + ad[d];
    v = (v > 0.0f) ? v : 0.2f * v;               // leaky_relu, slope 0.2
    esc[e] = v;
    atomicMaxF(&nmax[d], v);
}

__global__ void edge_den(const int* __restrict__ dst, const float* __restrict__ esc,
                         const float* __restrict__ nmax, float* __restrict__ den,
                         int nE, int nNodes)
{
    const int e = blockIdx.x * blockDim.x + threadIdx.x;
    if (e >= nE + nNodes) return;
    const int d = (e < nE) ? dst[e] : (e - nE);
    atomicAdd(&den[d], __expf(esc[e] - nmax[d]));
}

// One wave per edge: out[dst] += h[src] * softmax_coef. out pre-seeded with bias.
__global__ void edge_agg(const int* __restrict__ src, const int* __restrict__ dst,
                         const float* __restrict__ esc, const float* __restrict__ nmax,
                         const float* __restrict__ den, const float* __restrict__ h,
                         float* __restrict__ out, int nE, int nNodes, int D)
{
    const int wid  = blockIdx.x * (blockDim.x / WAVE) + threadIdx.x / WAVE;
    const int lane = threadIdx.x & (WAVE - 1);
    if (wid >= nE + nNodes) return;
    int s, d;
    if (wid < nE) { s = src[wid]; d = dst[wid]; } else { s = wid - nE; d = s; }
    const float coef = __expf(esc[wid] - nmax[d]) / den[d];
    const float* hs = h + (size_t)s * D;
    float* od = out + (size_t)d * D;
    for (int c = lane; c < D; c += WAVE)
        atomicAdd(&od[c], hs[c] * coef);
}

// ---------------------------------------------------------------------------
// Global mean pool
// ---------------------------------------------------------------------------
__global__ void pool_accum(const float* __restrict__ z, const int* __restrict__ batch,
                           float* __restrict__ gsum, float* __restrict__ gcnt,
                           int n, int D)
{
    const int wid  = blockIdx.x * (blockDim.x / WAVE) + threadIdx.x / WAVE;
    const int lane = threadIdx.x & (WAVE - 1);
    if (wid >= n) return;
    const int g = batch[wid];
    const float* zp = z + (size_t)wid * D;
    for (int c = lane; c < D; c += WAVE)
        atomicAdd(&gsum[(size_t)g * D + c], zp[c]);
    if (lane == 0) atomicAdd(&gcnt[g], 1.0f);
}

__global__ void pool_final(const float* __restrict__ gsum, const float* __restrict__ gcnt,
                           float* __restrict__ ge, int G, int D)
{
    const int i = blockIdx.x * blockDim.x + threadIdx.x;
    if (i >= G * D) return;
    ge[i] = gsum[i] / fmaxf(gcnt[i / D], 1.0f);
}

// ===========================================================================
// Host orchestration
// ===========================================================================
extern "C" void kernel_launch(void* const* d_in, const int* in_sizes, int n_in,
                              void* d_out, int out_size, void* d_ws, size_t ws_size,
                              hipStream_t stream)
{
    (void)in_sizes; (void)n_in; (void)out_size; (void)ws_size;
    const int Nn = 100000, E = 800000, G = 512;
    const int IN = 64, HID = 128, LAT = 64, OUT = 64;
    const int Etot = E + Nn;                     // edges + self loops

    const float* x    = (const float*)d_in[0];
    /* d_in[1] = edge_attr: ignored (edge_dim=None) */
    const float* W1   = (const float*)d_in[2];
    const float* a_s1 = (const float*)d_in[3];
    const float* a_d1 = (const float*)d_in[4];
    const float* b1   = (const float*)d_in[5];
    const float* W2   = (const float*)d_in[6];
    const float* a_s2 = (const float*)d_in[7];
    const float* a_d2 = (const float*)d_in[8];
    const float* b2   = (const float*)d_in[9];
    const float* Wl   = (const float*)d_in[10];
    const float* bl   = (const float*)d_in[11];
    const float* Wd1  = (const float*)d_in[12];
    const float* bd1  = (const float*)d_in[13];
    const float* Wd2  = (const float*)d_in[14];
    const float* bd2  = (const float*)d_in[15];
    const int*   ei   = (const int*)d_in[16];    // [2, E]
    const int*   bat  = (const int*)d_in[17];
    const int* srcA = ei;
    const int* dstA = ei + E;

    float* out_xhat = (float*)d_out;                     // [Nn, OUT]
    float* out_z    = out_xhat + (size_t)Nn * OUT;       // [Nn, LAT]
    float* out_ge   = out_z    + (size_t)Nn * LAT;       // [G,  LAT]

    // ---- workspace carve-out (floats) ----
    float* ws = (float*)d_ws;
    size_t o = 0;
    float* h1   = ws + o; o += (size_t)Nn * HID; // x@W1; later reused as decoder hidden
    float* hagg = ws + o; o += (size_t)Nn * HID; // layer-1 aggregated (relu'd)
    float* h2   = ws + o; o += (size_t)Nn * LAT; // hagg@W2
    float* zagg = ws + o; o += (size_t)Nn * LAT; // layer-2 aggregated
    float* as   = ws + o; o += (size_t)Nn;
    float* ad   = ws + o; o += (size_t)Nn;
    float* nmax = ws + o; o += (size_t)Nn;
    float* den  = ws + o; o += (size_t)Nn;
    float* esc  = ws + o; o += (size_t)Etot;
    float* gsum = ws + o; o += (size_t)G * LAT;
    float* gcnt = ws + o; o += (size_t)G;        // contiguous after gsum

    auto blocksT = [](long long t) { return (int)((t + 255) / 256); }; // thread/item
    auto blocksW = [](long long w) { return (int)((w + 7) / 8); };     // wave/item
    auto gemm = [&](const float* A, const float* B, const float* bias, float* C,
                    int M, int K, int Ncols, int act) {
        const int mblks = (M / 16 + 7) / 8;      // 8 M-tiles per block
        const int nblks = Ncols / BN;            // one 64-wide N panel per block
        gemm_wmma_f32<<<mblks * nblks, 256, 0, stream>>>(A, B, bias, C, M, K, Ncols, act);
    };

    // ===== GAT layer 1: h = relu(gat(x, W1) + b1) =====
    gemm(x, W1, nullptr, h1, Nn, IN, HID, 0);
    node_alpha<<<blocksW(Nn), 256, 0, stream>>>(h1, a_s1, a_d1, as, ad, Nn, HID);
    init_nodes<<<blocksT(Nn), 256, 0, stream>>>(nmax, den, Nn);
    bcast_bias<<<blocksT((long long)Nn * HID), 256, 0, stream>>>(hagg, b1, Nn * HID, HID);
    edge_max<<<blocksT(Etot), 256, 0, stream>>>(srcA, dstA, as, ad, esc, nmax, E, Nn);
    edge_den<<<blocksT(Etot), 256, 0, stream>>>(dstA, esc, nmax, den, E, Nn);
    edge_agg<<<blocksW(Etot), 256, 0, stream>>>(srcA, dstA, esc, nmax, den, h1, hagg, E, Nn, HID);
    relu_ip<<<blocksT((long long)Nn * HID), 256, 0, stream>>>(hagg, Nn * HID);

    // ===== GAT layer 2: z0 = gat(h, W2) + b2 =====
    gemm(hagg, W2, nullptr, h2, Nn, HID, LAT, 0);
    node_alpha<<<blocksW(Nn), 256, 0, stream>>>(h2, a_s2, a_d2, as, ad, Nn, LAT);
    init_nodes<<<blocksT(Nn), 256, 0, stream>>>(nmax, den, Nn);
    bcast_bias<<<blocksT((long long)Nn * LAT), 256, 0, stream>>>(zagg, b2, Nn * LAT, LAT);
    edge_max<<<blocksT(Etot), 256, 0, stream>>>(srcA, dstA, as, ad, esc, nmax, E, Nn);
    edge_den<<<blocksT(Etot), 256, 0, stream>>>(dstA, esc, nmax, den, E, Nn);
    edge_agg<<<blocksW(Etot), 256, 0, stream>>>(srcA, dstA, esc, nmax, den, h2, zagg, E, Nn, LAT);

    // ===== z = z0 @ Wl + bl =====
    gemm(zagg, Wl, bl, out_z, Nn, LAT, LAT, 0);

    // ===== global mean pool =====
    zero_f<<<blocksT((long long)G * LAT + G), 256, 0, stream>>>(gsum, G * LAT + G);
    pool_accum<<<blocksW(Nn), 256, 0, stream>>>(out_z, bat, gsum, gcnt, Nn, LAT);
    pool_final<<<blocksT((long long)G * LAT), 256, 0, stream>>>(gsum, gcnt, out_ge, G, LAT);

    // ===== decoder: x_hat = relu(z @ Wd1 + bd1) @ Wd2 + bd2 =====
    gemm(out_z, Wd1, bd1, h1, Nn, LAT, HID, 1);  // fused relu; reuses h1 buffer
    gemm(h1, Wd2, bd2, out_xhat, Nn, HID, OUT, 0);
}